// GraphConvolutionKeras_72430328480132
// MI455X (gfx1250) — compile-verified
//
#include <hip/hip_runtime.h>

#define N_NODES 100000
#define N_EDGES 1600000
#define N_SUP   2
#define IN_DIM  256
#define OUT_DIM 64

typedef __attribute__((ext_vector_type(2))) float v2f;
typedef __attribute__((ext_vector_type(8))) float v8f;

// ---------------------------------------------------------------------------
// Zero the output buffer (6.4M floats = 1.6M float4; 6250 blocks x 256 exact)
// ---------------------------------------------------------------------------
__global__ void __launch_bounds__(256) gcn_zero(float4* __restrict__ out) {
    const size_t i = (size_t)blockIdx.x * 256 + threadIdx.x;
    out[i] = make_float4(0.f, 0.f, 0.f, 0.f);
}

// ---------------------------------------------------------------------------
// GEMM: pre[sup] = x @ W[sup] with V_WMMA_F32_16X16X4_F32.
// Block = 256 threads = 8 wave32s; each wave owns a 16(M) x 64(N) strip
// (4 accumulator tiles). Per 64-row K-chunk:
//   - all 16 A fragments preloaded (global_load_b64, waits hoisted out of
//     the WMMA loop),
//   - W chunk staged in LDS *pre-swizzled into B-fragment order*: fragment
//     (k4,t) = 64 contiguous floats, lane L owns floats [2L, 2L+1]. Each B
//     operand is then a single even-aligned ds_load_b64 (no shuffle movs),
//     and the 256B-contiguous pattern is bank-conflict-free (lane L ->
//     banks 2L, 2L+1).
// ---------------------------------------------------------------------------
__global__ void __launch_bounds__(256)
gcn_gemm(const float* __restrict__ x, const float* __restrict__ W,
         float* __restrict__ pre) {
    constexpr int KC = 64;                 // K-chunk
    __shared__ float bsh[KC * OUT_DIM];    // 16 KB, WMMA-fragment order

    const int tid  = threadIdx.x;
    const int lane = tid & 31;
    const int wave = tid >> 5;
    const int sup  = blockIdx.y;

    const float* Wp   = W + (size_t)sup * IN_DIM * OUT_DIM;
    float*       prep = pre + (size_t)sup * N_NODES * OUT_DIM;

    int m_base = (blockIdx.x * 8 + wave) * 16;
    if (m_base > N_NODES - 16) m_base = N_NODES - 16;  // clamp; duplicates benign

    const int half = lane >> 4;   // 0: lanes 0-15, 1: lanes 16-31
    const int l16  = lane & 15;
    const float* xrow = x + (size_t)(m_base + l16) * IN_DIM + half * 2;

    // Per-thread constant part of the swizzled LDS fill:
    // f = tid + i*256  ->  k4 = i, j = tid&1, ln = (tid>>1)&31, t = (tid>>6)&3
    const int fill_j  = tid & 1;
    const int fill_ln = (tid >> 1) & 31;
    const int fill_t  = (tid >> 6) & 3;
    const int fill_k  = ((fill_ln >> 4) << 1) + fill_j;        // k offset within k4 group
    const int fill_n  = fill_t * 16 + (fill_ln & 15);          // column

    v8f acc[4] = {v8f{}, v8f{}, v8f{}, v8f{}};

    for (int kc = 0; kc < IN_DIM; kc += KC) {
        // --- Preload all A fragments for this chunk (16 x b64 per lane) ---
        v2f areg[16];
        #pragma unroll
        for (int ki = 0; ki < 16; ++ki)
            areg[ki] = *(const v2f*)(xrow + kc + ki * 4);

        // --- Fill LDS with W chunk in B-fragment order ---
        #pragma unroll
        for (int i = 0; i < 16; ++i) {
            bsh[tid + i * 256] =
                Wp[(size_t)(kc + i * 4 + fill_k) * OUT_DIM + fill_n];
        }
        __syncthreads();

        // --- 16 K-steps x 4 N-tiles; B = one ds_load_b64 per WMMA ---
        #pragma unroll
        for (int ki = 0; ki < 16; ++ki) {
            #pragma unroll
            for (int t = 0; t < 4; ++t) {
                const v2f b =
                    *(const v2f*)(bsh + (ki * 4 + t) * 64 + lane * 2);
                acc[t] = __builtin_amdgcn_wmma_f32_16x16x4_f32(
                    false, areg[ki], false, b, (short)0, acc[t], false, false);
            }
        }
        __syncthreads();
    }

    // Store D: VGPR v -> row m_base + v + half*8, col = t*16 + l16.
    #pragma unroll
    for (int v = 0; v < 8; ++v) {
        const size_t row = (size_t)(m_base + v + half * 8);
        float* op = prep + row * OUT_DIM + l16;
        op[0]  = acc[0][v];
        op[16] = acc[1][v];
        op[32] = acc[2][v];
        op[48] = acc[3][v];
    }
}

// ---------------------------------------------------------------------------
// SpMM scatter: 16 threads per edge, each handles 4 output dims.
// pre working set (51.2 MB) is L2-resident; unsafeAtomicAdd guarantees the
// hardware GLOBAL_ATOMIC_ADD_F32 (no-return / STOREcnt) path, no CAS loop.
// Grid: (E*16/256, N_SUP) = (100000, 2) exactly -> no bounds checks.
// ---------------------------------------------------------------------------
__global__ void __launch_bounds__(256)
gcn_scatter(const float* __restrict__ pre, const float* __restrict__ edge_val,
            const int* __restrict__ edge_src, const int* __restrict__ edge_dst,
            float* __restrict__ out) {
    const size_t gid = (size_t)blockIdx.x * 256 + threadIdx.x;
    const size_t e   = gid >> 4;            // edge index within support
    const int    q   = (int)(gid & 15) * 4; // first of 4 dims
    const int    sup = blockIdx.y;

    const size_t eoff = (size_t)sup * N_EDGES + e;
    const float v = edge_val[eoff];
    const int   s = edge_src[eoff];
    const int   d = edge_dst[eoff];

    const float4 p = *(const float4*)(pre + ((size_t)sup * N_NODES + (size_t)s) * OUT_DIM + q);
    float* o = out + (size_t)d * OUT_DIM + q;
    unsafeAtomicAdd(o + 0, p.x * v);
    unsafeAtomicAdd(o + 1, p.y * v);
    unsafeAtomicAdd(o + 2, p.z * v);
    unsafeAtomicAdd(o + 3, p.w * v);
}

// ---------------------------------------------------------------------------
// ReLU over the accumulated output.
// ---------------------------------------------------------------------------
__global__ void __launch_bounds__(256) gcn_relu(float4* __restrict__ out) {
    const size_t i = (size_t)blockIdx.x * 256 + threadIdx.x;
    float4 v = out[i];
    v.x = fmaxf(v.x, 0.f);
    v.y = fmaxf(v.y, 0.f);
    v.z = fmaxf(v.z, 0.f);
    v.w = fmaxf(v.w, 0.f);
    out[i] = v;
}

// ---------------------------------------------------------------------------
extern "C" void kernel_launch(void* const* d_in, const int* in_sizes, int n_in,
                              void* d_out, int out_size, void* d_ws, size_t ws_size,
                              hipStream_t stream) {
    (void)in_sizes; (void)n_in; (void)out_size; (void)ws_size;

    const float* x  = (const float*)d_in[0];  // [N_NODES, IN_DIM]
    const float* W  = (const float*)d_in[1];  // [N_SUP, IN_DIM, OUT_DIM]
    const float* ev = (const float*)d_in[2];  // [N_SUP, N_EDGES]
    const int*   es = (const int*)d_in[3];    // [N_SUP, N_EDGES]
    const int*   ed = (const int*)d_in[4];    // [N_SUP, N_EDGES]
    float* out = (float*)d_out;               // [N_NODES, OUT_DIM]
    float* pre = (float*)d_ws;                // [N_SUP, N_NODES, OUT_DIM] = 51.2 MB

    // 1) zero output (1,600,000 float4 -> 6250 x 256 exact)
    gcn_zero<<<6250, 256, 0, stream>>>((float4*)out);

    // 2) pre = x @ W[sup]; 6250 M-tiles / 8 waves -> 782 blocks, y = support
    dim3 ggrid(782, N_SUP);
    gcn_gemm<<<ggrid, 256, 0, stream>>>(x, W, pre);

    // 3) scatter-add; E*16 threads per support -> 100000 blocks, y = support
    dim3 sgrid(100000, N_SUP);
    gcn_scatter<<<sgrid, 256, 0, stream>>>(pre, ev, es, ed, out);

    // 4) relu
    gcn_relu<<<6250, 256, 0, stream>>>((float4*)out);
}